// DAS_1675037245581
// MI455X (gfx1250) — compile-verified
//
#include <hip/hip_runtime.h>
#include <stdint.h>

// ---- problem constants (match reference) ----------------------------------
constexpr int GNX  = 512;
constexpr int GNY  = 512;
constexpr int GNT  = 2048;    // time samples per trace
constexpr int GNS  = 128;     // sensors
constexpr int NBC  = 8;       // B(4) * channels(2)
constexpr int TILE = 16;      // 16x16 pixel tile per block (256 threads)
// Staged delay window per sensor, in samples. In-tile delay spread is
// <= ceil(sqrt(2)*15*DX/(VS*DT)) = 55, plus <=3 for 4-sample alignment and
// rounding slack -> 72 (multiple of 4 so b128 stays 16B aligned).
constexpr int WIN  = 72;
constexpr int NBUF = 3;       // triple-buffered ring -> one barrier/sensor

__global__ __launch_bounds__(256)
void das_kernel(const float* __restrict__ sd,   // (4,2,128,2048) f32
                const int*   __restrict__ sxy,  // (128,2) i32
                float*       __restrict__ out)  // (4,2,512,512) f32
{
    __shared__ __align__(16) float buf[NBUF][NBC * WIN];   // 3 x 2.25 KB

    const int tid = threadIdx.x;
    const int px0 = blockIdx.x * TILE;
    const int py0 = blockIdx.y * TILE;
    const int px  = px0 + (tid >> 4);   // grid-x pixel index (0-based)
    const int py  = py0 + (tid & 15);   // grid-y pixel index (0-based)

    float acc[NBC];
#pragma unroll
    for (int i = 0; i < NBC; ++i) acc[i] = 0.0f;

    // Block-uniform minimum delay index over the whole tile, aligned down to
    // a multiple of 4. Every float step is a monotone lower bound of the
    // per-pixel computation (same ops, smaller exact-integer inputs), so
    // t_lo <= t(pixel) for all pixels in the tile. Compiles to scalar-float.
    auto tile_t_lo = [&](int sx, int sy) -> int {
        int a1 = sx - px0 - (TILE - 1), b1 = sx - px0;
        int a2 = sy - py0 - (TILE - 1), b2 = sy - py0;
        int umin = max(max(a1, -b1), 0);
        int vmin = max(max(a2, -b2), 0);
        float du = (float)umin * 1e-4f;
        float dv = (float)vmin * 1e-4f;
        float dis = sqrtf(du * du + dv * dv);
        int t_lo = (int)(dis / 1550.0f / 2.5e-8f);
        return t_lo & ~3;
    };

    // Async-copy the [t_al, t_al+WIN) window of all 8 (b,c) planes of sensor
    // s into LDS ring slot `which` (compile-time const at every call site):
    // 8 planes x (WIN/4) x b128, one per lane.
    auto stage = [&](int s, int t_al, int which) {
        constexpr int QP = WIN / 4;               // float4 chunks per plane
        if (tid < NBC * QP) {
            int bc = tid / QP;
            int j  = tid - bc * QP;
            const float* g = sd + ((size_t)(bc * GNS + s)) * GNT + t_al + j * 4;
            uint32_t lds = (uint32_t)(uintptr_t)(&buf[which][bc * WIN + j * 4]);
            asm volatile("global_load_async_to_lds_b128 %0, %1, off"
                         :: "v"(lds), "v"((uint64_t)(uintptr_t)g)
                         : "memory");
        }
    };

    // 3-deep sensor pipeline state; all indices are compile-time constants
    // after the unroll below, so these live entirely in registers.
    int ssx[NBUF], ssy[NBUF], stal[NBUF];

    // Prologue: stage sensors 0 and 1 into slots 0 and 1.
#pragma unroll
    for (int k = 0; k < 2; ++k) {
        ssx[k]  = sxy[2 * k];
        ssy[k]  = sxy[2 * k + 1];
        stal[k] = tile_t_lo(ssx[k], ssy[k]);
        stage(k, stal[k], k);
    }

    // Sensor loop blocked by 3 so ring slots are constants: sensor sb+k uses
    // slot k and restages slot (k+2)%3 two sensors ahead.
    for (int sb = 0; sb < GNS; sb += NBUF) {
#pragma unroll
        for (int k = 0; k < NBUF; ++k) {
            const int s = sb + k;
            if (s < GNS) {                      // block-uniform guard
                // Own-wave async loads retire in order: <=1 outstanding means
                // the copy for sensor s has landed in LDS.
                if (s + 1 < GNS) {
                    asm volatile("s_wait_asynccnt 1" ::: "memory");
                } else {
                    asm volatile("s_wait_asynccnt 0" ::: "memory");
                }
                __syncthreads();  // staged data visible; reads of slot k2 done

                // Restage two ahead (slot k2 == (s-1)%3 was last read in
                // iteration s-1, which precedes this barrier for every wave).
                constexpr int k2 = 0;  // placeholder, real value below
                (void)k2;
                const int slot2 = (k + 2) % NBUF;   // compile-time constant
                if (s + 2 < GNS) {
                    int sx = sxy[2 * (s + 2)];
                    int sy = sxy[2 * (s + 2) + 1];
                    int ta = tile_t_lo(sx, sy);
                    ssx[slot2] = sx; ssy[slot2] = sy; stal[slot2] = ta;
                    stage(s + 2, ta, slot2);
                }

                // Per-pixel delay, replicating reference float semantics:
                // (x - ix + 1)*DX with ix = px+1 == (x - px)*DX, exact diff.
                float dxp = (float)(ssx[k] - px) * 1e-4f;
                float dyp = (float)(ssy[k] - py) * 1e-4f;
                float dis = sqrtf(dxp * dxp + dyp * dyp);
                int t  = (int)(dis / 1550.0f / 2.5e-8f);
                int dt = t - stal[k];          // in [0, WIN)

                const float* b = &buf[k][0];   // constant LDS base
#pragma unroll
                for (int bc = 0; bc < NBC; ++bc)
                    acc[bc] += b[bc * WIN + dt];   // 8x ds_load_b32
            }
        }
    }

    const size_t p = (size_t)px * GNY + py;
#pragma unroll
    for (int bc = 0; bc < NBC; ++bc)
        out[(size_t)bc * (GNX * GNY) + p] = acc[bc];
}

extern "C" void kernel_launch(void* const* d_in, const int* in_sizes, int n_in,
                              void* d_out, int out_size, void* d_ws, size_t ws_size,
                              hipStream_t stream) {
    (void)in_sizes; (void)n_in; (void)out_size; (void)d_ws; (void)ws_size;
    const float* sd  = (const float*)d_in[0];   // sensor_data (4,2,128,2048)
    const int*   sxy = (const int*)d_in[1];     // sensor_xy   (128,2)
    float*       out = (float*)d_out;           // (4,2,512,512)
    dim3 grid(GNX / TILE, GNY / TILE);
    das_kernel<<<grid, dim3(256), 0, stream>>>(sd, sxy, out);
}